// MegaMerge_15539191677616
// MI455X (gfx1250) — compile-verified
//
#include <hip/hip_runtime.h>
#include <cstdint>

// ---------------------------------------------------------------------------
// MegaMerge: G = concat([h^T, c2q, h^T*c2q, h^T*q2c], axis=0)
//   h   : [1, T, D] fp32 (row-major, d fastest)
//   c2q : [D, T]    fp32 (row-major, t fastest)
//   q2c : [D, T]    fp32
//   G   : [4D, T]   fp32
//
// Pure bandwidth problem (~224 MiB min traffic, ~10 us @ 23.3 TB/s).
// h^T is produced via CDNA5 async global->LDS DMA with per-lane LDS
// destination addresses performing the transpose on write. All other global
// traffic is b128-coalesced. T/D are compile-time constants so all addressing
// is 32-bit shift/or arithmetic (no v_mul_u64 chains).
// ---------------------------------------------------------------------------

constexpr uint32_t kT = 4096;   // max_context_length (reference setup)
constexpr uint32_t kD = 2048;

#define TT    32   // tile extent along t (contiguous dim of c2q/q2c/out rows)
#define TD    32   // tile extent along d
#define PITCH 37   // LDS row pitch in dwords; gcd(37,64)=1 -> conflict-free

typedef __attribute__((address_space(1))) int  glob_i32_t;  // "__device__ int"
typedef __attribute__((address_space(3))) int  lds_i32_t;   // "__shared__ int"
typedef __attribute__((address_space(3))) char lds_char_t;

__device__ __forceinline__ void async_copy_dword_to_lds(const float* g, float* l) {
#if defined(__has_builtin) && __has_builtin(__builtin_amdgcn_global_load_async_to_lds_b32)
    // Signature (per hipcc diagnostic): (AS1 int* gsrc, AS3 int* ldst, Ii offset, Ii cpol)
    __builtin_amdgcn_global_load_async_to_lds_b32(
        (glob_i32_t*)g, (lds_i32_t*)l, 0, 0);
#else
    // Portable fallback per CDNA5 ISA: GV mode, per-lane
    // 64-bit global address in a VGPR pair, per-lane LDS byte offset in VDST.
    const uint32_t lds_off = (uint32_t)(uintptr_t)(lds_char_t*)l;
    const uint64_t gaddr   = (uint64_t)(uintptr_t)g;
    asm volatile("global_load_async_to_lds_b32 %0, %1, off"
                 :: "v"(lds_off), "v"(gaddr)
                 : "memory");
#endif
}

__device__ __forceinline__ void wait_async_zero() {
    asm volatile("s_wait_asynccnt 0" ::: "memory");
}

__global__ __launch_bounds__(256) void megamerge_kernel(
    const float* __restrict__ h,    // [T, D]
    const float* __restrict__ c2q,  // [D, T]
    const float* __restrict__ q2c,  // [D, T]
    float* __restrict__ out)        // [4D, T]
{
    __shared__ float lds_h[TD * PITCH];   // lds_h[d_local][t_local], pitch 37

    const uint32_t t0   = blockIdx.x * TT;
    const uint32_t d0   = blockIdx.y * TD;
    const uint32_t tid  = threadIdx.x;
    const uint32_t lane = tid & 31u;    // d_local for the DMA phase
    const uint32_t wave = tid >> 5;     // 0..7

    // ---- Phase 1: async DMA h tile -> LDS, transposed on write -------------
    // Wave w, iteration i: lanes read 128 contiguous bytes of row t0+4w+i
    // (cols d0..d0+31) and scatter them down LDS column t_local = 4w+i.
    #pragma unroll
    for (uint32_t i = 0; i < 4; ++i) {
        const uint32_t tl = wave * 4u + i;                    // t_local 0..31
        const float*   g  = h + ((t0 + tl) * kD + (d0 + lane));  // 32-bit idx
        float*         l  = &lds_h[lane * PITCH + tl];
        async_copy_dword_to_lds(g, l);
    }
    wait_async_zero();   // this wave's DMA data landed in LDS
    __syncthreads();     // make all waves' tiles visible

    // ---- Phase 2: compute one float4 along t per thread --------------------
    const uint32_t dl  = tid >> 3;           // d_local 0..31
    const uint32_t tl4 = (tid & 7u) * 4u;    // t_local 0,4,...,28

    const float hv0 = lds_h[dl * PITCH + tl4 + 0];
    const float hv1 = lds_h[dl * PITCH + tl4 + 1];
    const float hv2 = lds_h[dl * PITCH + tl4 + 2];
    const float hv3 = lds_h[dl * PITCH + tl4 + 3];

    const uint32_t base = (d0 + dl) * kT + (t0 + tl4);   // < 2^23, 16B aligned
    constexpr uint32_t DT = kD * kT;                     // 8,388,608 elements

    const float4 c = *(const float4*)(c2q + base);
    const float4 q = *(const float4*)(q2c + base);

    float4 hb; hb.x = hv0; hb.y = hv1; hb.z = hv2; hb.w = hv3;
    float4 hc; hc.x = hv0 * c.x; hc.y = hv1 * c.y; hc.z = hv2 * c.z; hc.w = hv3 * c.w;
    float4 hq; hq.x = hv0 * q.x; hq.y = hv1 * q.y; hq.z = hv2 * q.z; hq.w = hv3 * q.w;

    *(float4*)(out +           base) = hb;   // block 0: h^T
    *(float4*)(out + DT      + base) = c;    // block 1: c2q
    *(float4*)(out + 2u * DT + base) = hc;   // block 2: h^T * c2q
    *(float4*)(out + 3u * DT + base) = hq;   // block 3: h^T * q2c
}

extern "C" void kernel_launch(void* const* d_in, const int* in_sizes, int n_in,
                              void* d_out, int out_size, void* d_ws, size_t ws_size,
                              hipStream_t stream) {
    (void)in_sizes; (void)n_in; (void)out_size; (void)d_ws; (void)ws_size;

    const float* h   = (const float*)d_in[0];
    const float* c2q = (const float*)d_in[1];
    const float* q2c = (const float*)d_in[2];
    float*       out = (float*)d_out;

    dim3 grid(kT / TT, kD / TD);   // 128 x 64 = 8192 blocks
    dim3 block(256);
    megamerge_kernel<<<grid, block, 0, stream>>>(h, c2q, q2c, out);
}